// MetaphorSimilarityModel_86930138071227
// MI455X (gfx1250) — compile-verified
//
#include <hip/hip_runtime.h>

typedef float v2f __attribute__((ext_vector_type(2)));
typedef float v8f __attribute__((ext_vector_type(8)));

#define M_CAND 256
#define D_DIM  1024
#define EPSF   1e-8f

__launch_bounds__(256, 2)
__global__ void metaphor_sim_topk_kernel(const float* __restrict__ query,
                                         const float* __restrict__ bank,
                                         const float* __restrict__ bank_labels,
                                         const int*   __restrict__ bucket_idx,
                                         const int*   __restrict__ kptr,
                                         float*       __restrict__ out)
{
    __shared__ float s_q[D_DIM];        // staged query row
    __shared__ float s_sims[M_CAND];
    __shared__ float s_lab[M_CAND];
    __shared__ float s_qnorm2;
    __shared__ float s_rv[256];
    __shared__ int   s_ri[256];

    const int s    = blockIdx.x;
    const int tid  = threadIdx.x;
    const int lane = tid & 31;
    const int wave = tid >> 5;

    if (tid == 0) s_qnorm2 = 0.0f;
    __syncthreads();

    // ---- stage query row into LDS (float4 per thread) + block norm^2 ----
    {
        const float4* q4 = (const float4*)(query + (size_t)s * D_DIM);
        float4 v = q4[tid];                       // 256 threads * 4 floats = 1024
        ((float4*)s_q)[tid] = v;
        float sq = v.x * v.x + v.y * v.y + v.z * v.z + v.w * v.w;
        #pragma unroll
        for (int off = 16; off > 0; off >>= 1)
            sq += __shfl_down(sq, off, 32);
        if (lane == 0) atomicAdd(&s_qnorm2, sq);  // ds_add_f32
    }
    __syncthreads();

    // ---- per-wave: 32 candidates as two 16-wide WMMA column groups ----
    const int h  = lane >> 4;     // K-half within the 4-wide chunk
    const int nn = lane & 15;     // WMMA column (candidate within group)

    const int m0 = wave * 32 + nn;       // group 0 candidate
    const int m1 = m0 + 16;              // group 1 candidate
    const long long i0 = (long long)bucket_idx[s * M_CAND + m0];
    const long long i1 = (long long)bucket_idx[s * M_CAND + m1];
    const float* p0 = bank + (size_t)i0 * D_DIM + 2 * h;
    const float* p1 = bank + (size_t)i1 * D_DIM + 2 * h;

    v8f acc0 = {};
    v8f acc1 = {};
    float cn0 = 0.0f, cn1 = 0.0f;

    #pragma unroll 4
    for (int k = 0; k < D_DIM; k += 4) {
        // A fragment: query chunk, replicated over all 16 rows.
        // lanes 0-15 hold K={k,k+1}; lanes 16-31 hold K={k+2,k+3}.
        v2f a = *(const v2f*)(s_q + k + 2 * h);          // ds_load_b64
        // B fragments: column nn = candidate's dims (same K split as A)
        v2f b0 = *(const v2f*)(p0 + k);                  // global_load_b64 gather
        v2f b1 = *(const v2f*)(p1 + k);
        cn0 += b0.x * b0.x + b0.y * b0.y;
        cn1 += b1.x * b1.x + b1.y * b1.y;
        acc0 = __builtin_amdgcn_wmma_f32_16x16x4_f32(false, a, false, b0,
                                                     (short)0, acc0, false, false);
        acc1 = __builtin_amdgcn_wmma_f32_16x16x4_f32(false, a, false, b1,
                                                     (short)0, acc1, false, false);
    }

    // combine the two K-halves of each candidate's norm^2
    cn0 += __shfl_xor(cn0, 16, 32);
    cn1 += __shfl_xor(cn1, 16, 32);

    const float qn = fmaxf(sqrtf(s_qnorm2), EPSF);
    if (lane < 16) {
        // all A rows identical -> acc[0] on lane nn is dot(q, cand)
        float d0 = acc0[0];
        float d1 = acc1[0];
        s_sims[m0] = d0 / (qn * fmaxf(sqrtf(cn0), EPSF));
        s_sims[m1] = d1 / (qn * fmaxf(sqrtf(cn1), EPSF));
        s_lab[m0]  = bank_labels[i0];
        s_lab[m1]  = bank_labels[i1];
    }
    __syncthreads();

    // ---- top-k (k rounds of block-wide argmax, tie -> lower index) ----
    int kv = kptr[0];
    if (kv < 1) kv = 1;
    if (kv > M_CAND) kv = M_CAND;

    float lab_sum = 0.0f;  // only thread 0's copy is used
    for (int r = 0; r < kv; ++r) {
        s_rv[tid] = s_sims[tid];
        s_ri[tid] = tid;
        __syncthreads();
        #pragma unroll
        for (int stride = 128; stride > 0; stride >>= 1) {
            if (tid < stride) {
                float ov = s_rv[tid + stride];
                int   oi = s_ri[tid + stride];
                float mv = s_rv[tid];
                int   mi = s_ri[tid];
                if (ov > mv || (ov == mv && oi < mi)) {
                    s_rv[tid] = ov;
                    s_ri[tid] = oi;
                }
            }
            __syncthreads();
        }
        int best = s_ri[0];
        if (tid == 0) {
            lab_sum += s_lab[best];
            s_sims[best] = -3.0e38f;   // remove from next round
        }
        __syncthreads();
    }

    if (tid == 0)
        out[s] = rintf(lab_sum / (float)kv);   // round-half-even == jnp.round
}

extern "C" void kernel_launch(void* const* d_in, const int* in_sizes, int n_in,
                              void* d_out, int out_size, void* d_ws, size_t ws_size,
                              hipStream_t stream)
{
    const float* query       = (const float*)d_in[0];   // [S, 1024]
    const float* bank        = (const float*)d_in[1];   // [N, 1024]
    const float* bank_labels = (const float*)d_in[2];   // [N]
    const int*   bucket_idx  = (const int*)d_in[3];     // [S, 256]
    const int*   kptr        = (const int*)d_in[4];     // scalar k
    float*       out         = (float*)d_out;           // [S]

    const int S = out_size;  // one block per query token
    hipLaunchKernelGGL(metaphor_sim_topk_kernel, dim3(S), dim3(256), 0, stream,
                       query, bank, bank_labels, bucket_idx, kptr, out);
}